// GCN_51049981280538
// MI455X (gfx1250) — compile-verified
//
#include <hip/hip_runtime.h>
#include <hip/hip_bf16.h>

// ---------------------------------------------------------------------------
// 2-layer GCN for MI455X (gfx1250).
// Dense transforms use V_WMMA_F32_16X16X4_F32 (exact fp32 path).
// Edge aggregation is float4-vectorized gather + global_atomic_add_f32 scatter;
// working set (~26 MB) fits in the 192 MB L2, so atomics stay on-chip.
// ---------------------------------------------------------------------------

#define NNODES 50000
#define NEDGES 800000
#define F      64          // F_IN == F_HID == 64

typedef __attribute__((ext_vector_type(2))) float v2f;
typedef __attribute__((ext_vector_type(8))) float v8f;

// ---------------- degree / normalization ----------------

__global__ __launch_bounds__(256) void k_init_deg(float* __restrict__ deg) {
    int i = blockIdx.x * 256 + threadIdx.x;
    if (i < NNODES) deg[i] = 1.0f;               // self-loop contributes 1
}

__global__ __launch_bounds__(256) void k_count_deg(const int* __restrict__ dst,
                                                   float* __restrict__ deg) {
    int e = blockIdx.x * 256 + threadIdx.x;
    if (e < NEDGES) atomicAdd(&deg[dst[e]], 1.0f);
}

__global__ __launch_bounds__(256) void k_rsqrt(float* __restrict__ deg) {
    int i = blockIdx.x * 256 + threadIdx.x;
    if (i < NNODES) deg[i] = rsqrtf(deg[i]);     // deg >= 1 always
}

// ---------------- dense transform: H = X @ W via WMMA fp32 ----------------
// One wave computes a 16x16 tile of H. Block = 128 threads = 4 waves, each
// wave owns one of the 4 column tiles (F=64). blockIdx.x = row tile (3125).

__global__ __launch_bounds__(128) void k_gemm_wmma(const float* __restrict__ X,
                                                   const float* __restrict__ W,
                                                   float* __restrict__ H) {
    const int wave = threadIdx.x >> 5;      // 0..3  -> N tile
    const int lane = threadIdx.x & 31;
    const int half = lane >> 4;             // 0: lanes 0-15, 1: lanes 16-31
    const int l16  = lane & 15;
    const int mrow = blockIdx.x * 16;       // 3125 * 16 = 50000 exactly
    const int ncol = wave * 16 + l16;

    const float* __restrict__ xrow = X + (size_t)(mrow + l16) * F;

    v8f c = {};
#pragma unroll
    for (int k = 0; k < F; k += 4) {
        const int kk = k + 2 * half;
        // A 16x4: VGPR0 = K {0|2}, VGPR1 = K {1|3} per lane half
        v2f a;
        a.x = xrow[kk];
        a.y = xrow[kk + 1];
        // B 4x16: rows striped across lanes within each VGPR
        v2f b;
        b.x = W[(size_t)kk * F + ncol];
        b.y = W[(size_t)(kk + 1) * F + ncol];
        c = __builtin_amdgcn_wmma_f32_16x16x4_f32(
                /*neg_a=*/false, a, /*neg_b=*/false, b,
                /*c_mod=*/(short)0, c, /*reuse_a=*/false, /*reuse_b=*/false);
    }

    // D layout: VGPR r -> row (r + 8*half), col = l16
    float* __restrict__ o = H + (size_t)(mrow + 8 * half) * F + ncol;
#pragma unroll
    for (int r = 0; r < 8; ++r) o[(size_t)r * F] = c[r];
}

// ---------------- zero the aggregation buffer ----------------

__global__ __launch_bounds__(256) void k_zero4(float4* __restrict__ p, int n4) {
    int i = blockIdx.x * 256 + threadIdx.x;
    if (i < n4) p[i] = make_float4(0.f, 0.f, 0.f, 0.f);
}

// ---------------- edge scatter: agg[dst] += h[src] * dinv[src]*dinv[dst] ----
// 16 threads per edge, one float4 (4 features) each.

__global__ __launch_bounds__(256) void k_scatter(const int*   __restrict__ src,
                                                 const int*   __restrict__ dst,
                                                 const float* __restrict__ dinv,
                                                 const float* __restrict__ H,
                                                 float*       __restrict__ agg) {
    long long gid = (long long)blockIdx.x * 256 + threadIdx.x;
    int e  = (int)(gid >> 4);
    int c4 = (int)(gid & 15);
    if (e >= NEDGES) return;
    int s = src[e];
    int d = dst[e];
    float nrm = dinv[s] * dinv[d];
    float4 v = ((const float4*)(H + (size_t)s * F))[c4];
    float* o = agg + (size_t)d * F + (size_t)c4 * 4;
    atomicAdd(o + 0, v.x * nrm);
    atomicAdd(o + 1, v.y * nrm);
    atomicAdd(o + 2, v.z * nrm);
    atomicAdd(o + 3, v.w * nrm);
}

// ---------------- finalize: out = relu(agg + h*dinv^2 + b) ----------------
// Folds the self-loop message in without atomics.

__global__ __launch_bounds__(256) void k_finalize(const float* __restrict__ agg,
                                                  const float* __restrict__ H,
                                                  const float* __restrict__ dinv,
                                                  const float* __restrict__ bias,
                                                  float*       __restrict__ out) {
    int gid = blockIdx.x * 256 + threadIdx.x;   // node*16 + chunk
    int i  = gid >> 4;
    int c4 = gid & 15;
    if (i >= NNODES) return;
    float di = dinv[i];
    float sl = di * di;
    float4 a = ((const float4*)(agg + (size_t)i * F))[c4];
    float4 h = ((const float4*)(H   + (size_t)i * F))[c4];
    float4 b = ((const float4*)bias)[c4];
    float4 r;
    r.x = fmaxf(fmaf(h.x, sl, a.x) + b.x, 0.f);
    r.y = fmaxf(fmaf(h.y, sl, a.y) + b.y, 0.f);
    r.z = fmaxf(fmaf(h.z, sl, a.z) + b.z, 0.f);
    r.w = fmaxf(fmaf(h.w, sl, a.w) + b.w, 0.f);
    ((float4*)(out + (size_t)i * F))[c4] = r;
}

// ---------------------------------------------------------------------------

extern "C" void kernel_launch(void* const* d_in, const int* in_sizes, int n_in,
                              void* d_out, int out_size, void* d_ws, size_t ws_size,
                              hipStream_t stream) {
    const float* x   = (const float*)d_in[0];
    const int*   ei  = (const int*)  d_in[1];   // [2, NEDGES]
    const float* W1  = (const float*)d_in[2];
    const float* b1  = (const float*)d_in[3];
    const float* W2  = (const float*)d_in[4];
    const float* b2  = (const float*)d_in[5];
    float*       out = (float*)d_out;

    const int* src = ei;            // edge_index[0]
    const int* dst = ei + NEDGES;   // edge_index[1]

    // workspace layout (floats): dinv | h | y | agg
    float* ws   = (float*)d_ws;
    float* dinv = ws;
    float* h    = dinv + NNODES;
    float* y    = h    + (size_t)NNODES * F;
    float* agg  = y    + (size_t)NNODES * F;

    const int gNode  = (NNODES + 255) / 256;          // 196
    const int gEdge  = (NEDGES + 255) / 256;          // 3125
    const int gGemm  = NNODES / 16;                   // 3125 blocks x 128 thr
    const int nAgg4  = NNODES * (F / 4);              // 800000 float4
    const int gZero  = (nAgg4 + 255) / 256;           // 3125
    const int gScat  = (NEDGES * 16 + 255) / 256;     // 50000
    const int gFin   = (NNODES * 16 + 255) / 256;     // 3125

    // degree normalization
    k_init_deg <<<gNode, 256, 0, stream>>>(dinv);
    k_count_deg<<<gEdge, 256, 0, stream>>>(dst, dinv);
    k_rsqrt    <<<gNode, 256, 0, stream>>>(dinv);

    // ---- layer 1 ----
    k_gemm_wmma<<<gGemm, 128, 0, stream>>>(x, W1, h);
    k_zero4    <<<gZero, 256, 0, stream>>>((float4*)agg, nAgg4);
    k_scatter  <<<gScat, 256, 0, stream>>>(src, dst, dinv, h, agg);
    k_finalize <<<gFin,  256, 0, stream>>>(agg, h, dinv, b1, y);

    // ---- layer 2 ----
    k_gemm_wmma<<<gGemm, 128, 0, stream>>>(y, W2, h);
    k_zero4    <<<gZero, 256, 0, stream>>>((float4*)agg, nAgg4);
    k_scatter  <<<gScat, 256, 0, stream>>>(src, dst, dinv, h, agg);
    k_finalize <<<gFin,  256, 0, stream>>>(agg, h, dinv, b2, out);
}